// pde_36155034698048
// MI455X (gfx1250) — compile-verified
//
#include <hip/hip_runtime.h>
#include <hip/hip_bf16.h>
#include <math.h>

// Problem constants (fixed by the reference)
#define BB 32
#define HH 768
#define WW 768
static constexpr float EPS_CURV = 1e-8f;
static constexpr float EPS_BN   = 1e-5f;

// Tiling for the fused pass
static constexpr int TW    = 128;         // tile width (pixels)
static constexpr int TH    = 16;          // tile height
static constexpr int LST   = 132;         // LDS row stride: 528B = 16B-aligned rows
static constexpr int LROWS = TH + 2;      // rows incl. halo (18)
static constexpr int NTHREADS = 256;      // 8 waves (wave32)
static constexpr int NB_RED = 2048;       // blocks for the reduction pass
// LDS row layout: [0..127] = interior cols tx0..tx0+127 (16B aligned),
//                 [128] = left halo col (tx0-1), [129] = right halo col (tx0+128),
//                 [130..131] = alignment padding (unused)

// ---------------- CDNA5 async global->LDS path (guarded) ----------------
#if defined(__gfx1250__) && __has_builtin(__builtin_amdgcn_global_load_async_to_lds_b32)
#define HAVE_ASYNC_LDS 1
#endif

typedef int v4i_t __attribute__((ext_vector_type(4)));

__device__ __forceinline__ void async_g2l_f32(const float* g, float* l) {
#ifdef HAVE_ASYNC_LDS
    __builtin_amdgcn_global_load_async_to_lds_b32(
        (__attribute__((address_space(1))) int*)g,
        (__attribute__((address_space(3))) int*)l,
        0, 0);
#else
    *l = *g;
#endif
}

__device__ __forceinline__ void async_g2l_f32x4(const float* g, float* l) {
#if defined(HAVE_ASYNC_LDS) && __has_builtin(__builtin_amdgcn_global_load_async_to_lds_b128)
    __builtin_amdgcn_global_load_async_to_lds_b128(
        (__attribute__((address_space(1))) v4i_t*)g,
        (__attribute__((address_space(3))) v4i_t*)l,
        0, 0);
#elif defined(HAVE_ASYNC_LDS)
    async_g2l_f32(g + 0, l + 0);
    async_g2l_f32(g + 1, l + 1);
    async_g2l_f32(g + 2, l + 2);
    async_g2l_f32(g + 3, l + 3);
#else
    l[0] = g[0]; l[1] = g[1]; l[2] = g[2]; l[3] = g[3];
#endif
}

__device__ __forceinline__ void async_wait_all() {
#ifdef HAVE_ASYNC_LDS
#if __has_builtin(__builtin_amdgcn_s_wait_asynccnt)
    __builtin_amdgcn_s_wait_asynccnt(0);
#else
    asm volatile("s_wait_asynccnt 0" ::: "memory");
#endif
#endif
}

// ---------------- Pass 1: conv(h) + deterministic sum / sumsq ----------------
__global__ __launch_bounds__(NTHREADS)
void hsum_kernel(const float* __restrict__ x, const float* __restrict__ f,
                 const float* __restrict__ wconv,
                 float* __restrict__ psum, float* __restrict__ psqr)
{
    float w[18];
#pragma unroll
    for (int i = 0; i < 18; ++i) w[i] = wconv[i];

    const int total  = BB * HH * WW;            // 18,874,368 < 2^31
    const int stride = gridDim.x * blockDim.x;

    float s1 = 0.f, s2 = 0.f;
    for (int p = blockIdx.x * blockDim.x + threadIdx.x; p < total; p += stride) {
        int b   = p / (HH * WW);
        int rem = p - b * (HH * WW);
        int y   = rem / WW;
        int c   = rem - y * WW;
        const float* xb = x + (size_t)b * HH * WW;
        const float* fb = f + (size_t)b * HH * WW;

        float h = 0.f;
#pragma unroll
        for (int ky = 0; ky < 3; ++ky) {
            int yy = y + ky - 1;
            bool yok = (yy >= 0) & (yy < HH);
#pragma unroll
            for (int kx = 0; kx < 3; ++kx) {
                int xx = c + kx - 1;
                bool ok = yok & (xx >= 0) & (xx < WW);
                int idx = yy * WW + xx;
                float xv = ok ? xb[idx] : 0.f;
                float fv = ok ? fb[idx] : 0.f;
                h = fmaf(xv, w[ky * 3 + kx], h);
                h = fmaf(fv, w[9 + ky * 3 + kx], h);
            }
        }
        s1 += h;
        s2 = fmaf(h, h, s2);
    }

    // wave32 tree reduction, then cross-wave via LDS (deterministic)
    __shared__ float rs[NTHREADS / 32], rq[NTHREADS / 32];
#pragma unroll
    for (int off = 16; off > 0; off >>= 1) {
        s1 += __shfl_down(s1, off);
        s2 += __shfl_down(s2, off);
    }
    int lane = threadIdx.x & 31, wid = threadIdx.x >> 5;
    if (lane == 0) { rs[wid] = s1; rq[wid] = s2; }
    __syncthreads();
    if (wid == 0) {
        s1 = (lane < NTHREADS / 32) ? rs[lane] : 0.f;
        s2 = (lane < NTHREADS / 32) ? rq[lane] : 0.f;
#pragma unroll
        for (int off = 4; off > 0; off >>= 1) {
            s1 += __shfl_down(s1, off);
            s2 += __shfl_down(s2, off);
        }
        if (lane == 0) { psum[blockIdx.x] = s1; psqr[blockIdx.x] = s2; }
    }
}

// ---------------- Pass 1b: fold partials -> BN scale/shift ----------------
__global__ __launch_bounds__(NTHREADS)
void bnstats_kernel(const float* __restrict__ psum, const float* __restrict__ psqr,
                    const float* __restrict__ gamma, const float* __restrict__ beta,
                    float* __restrict__ bnp, int nb)
{
    float s1 = 0.f, s2 = 0.f;
    for (int i = threadIdx.x; i < nb; i += blockDim.x) { s1 += psum[i]; s2 += psqr[i]; }
    __shared__ float rs[NTHREADS / 32], rq[NTHREADS / 32];
#pragma unroll
    for (int off = 16; off > 0; off >>= 1) {
        s1 += __shfl_down(s1, off);
        s2 += __shfl_down(s2, off);
    }
    int lane = threadIdx.x & 31, wid = threadIdx.x >> 5;
    if (lane == 0) { rs[wid] = s1; rq[wid] = s2; }
    __syncthreads();
    if (threadIdx.x == 0) {
        s1 = 0.f; s2 = 0.f;
#pragma unroll
        for (int i = 0; i < NTHREADS / 32; ++i) { s1 += rs[i]; s2 += rq[i]; }
        const float n   = (float)((long)BB * HH * WW);
        float mu    = s1 / n;
        float var   = s2 / n - mu * mu;
        float inv   = 1.0f / sqrtf(var + EPS_BN);
        float scale = gamma[0] * inv;
        float shift = beta[0] - mu * scale;
        bnp[0] = scale;
        bnp[1] = shift;
    }
}

// ---------------- Pass 2: fused stencil + conv + BN + sigmoid ----------------
__global__ __launch_bounds__(NTHREADS)
void fused_kernel(const float* __restrict__ x, const float* __restrict__ f,
                  const float* __restrict__ wconv, const float* __restrict__ bnp,
                  float* __restrict__ out)
{
    __shared__ float xs[LROWS * LST];
    __shared__ float fs[LROWS * LST];

    const int tilesX = (WW + TW - 1) / TW;
    const int tx0 = (blockIdx.x % tilesX) * TW;
    const int ty0 = (blockIdx.x / tilesX) * TH;
    const int b   = blockIdx.y;
    const float* xb = x + (size_t)b * HH * WW;
    const float* fb = f + (size_t)b * HH * WW;
    const int tid = threadIdx.x;

    float w[18];
#pragma unroll
    for (int i = 0; i < 18; ++i) w[i] = wconv[i];
    const float scale = bnp[0], shift = bnp[1];

    // Interior tiles have a fully in-range halo -> skip zero-fill (uniform branch)
    const bool interior = (ty0 >= 1) && (tx0 >= 1) &&
                          (ty0 + TH + 1 <= HH) && (tx0 + TW + 1 <= WW);
    if (!interior) {
        for (int i = tid; i < LROWS * LST; i += NTHREADS) { xs[i] = 0.f; fs[i] = 0.f; }
        __syncthreads();
    }

    const int gy0 = ty0 - 1;

    // Interior: 18 rows x 32 b128 chunks per array (global + LDS both 16B aligned)
    for (int i = tid; i < LROWS * (TW / 4); i += NTHREADS) {
        int r = i >> 5;          // row 0..17
        int j = i & 31;          // chunk 0..31
        int gy = gy0 + r;
        if (gy >= 0 && gy < HH) {
            size_t g = (size_t)gy * WW + tx0 + 4 * j;
            async_g2l_f32x4(xb + g, &xs[r * LST + 4 * j]);
            async_g2l_f32x4(fb + g, &fs[r * LST + 4 * j]);
        }
    }
    // Halo columns: left -> LDS col 128, right -> LDS col 129 (b32 each)
    if (tid < LROWS * 2) {
        int r    = tid >> 1;
        int side = tid & 1;
        int gy = gy0 + r;
        int gx = side ? (tx0 + TW) : (tx0 - 1);
        if (gy >= 0 && gy < HH && gx >= 0 && gx < WW) {
            size_t g = (size_t)gy * WW + gx;
            async_g2l_f32(xb + g, &xs[r * LST + TW + side]);
            async_g2l_f32(fb + g, &fs[r * LST + TW + side]);
        }
    }
    async_wait_all();
    __syncthreads();

    // Each thread: 8 output pixels, column lx, rows ry, ry+2, ..., ry+14
    const int lx = tid & (TW - 1);
    const int ry = tid >> 7;
    const int cL = (lx == 0)      ? TW     : lx - 1;   // left neighbor column in LDS
    const int cR = (lx == TW - 1) ? TW + 1 : lx + 1;   // right neighbor column in LDS
#pragma unroll
    for (int k = 0; k < 8; ++k) {
        int orow = ry + 2 * k;
        int gy = ty0 + orow, gx = tx0 + lx;
        if (gy >= HH || gx >= WW) continue;

        const float* xr0 = &xs[orow * LST];        // LDS row of gy-1
        const float* xr1 = xr0 + LST;              // LDS row of gy
        const float* xr2 = xr1 + LST;              // LDS row of gy+1
        float ul = xr0[cL], up = xr0[lx], ur = xr0[cR];
        float lf = xr1[cL], v  = xr1[lx], rt = xr1[cR];
        float dl = xr2[cL], dn = xr2[lx], dr = xr2[cR];

        float d02 = up - v,  d04 = rt - v,  d06 = dn - v,  d08 = lf - v;
        float d13 = ur - ul, d53 = ur - dr, d62 = up - dn;
        float d71 = ul - dl, d75 = dr - dl, d84 = rt - lf;

        float c02 = 1.f / (EPS_CURV + sqrtf(fmaf(d02, d02, (d13 * d13 + d84 * d84) * 0.125f)));
        float c06 = 1.f / (EPS_CURV + sqrtf(fmaf(d06, d06, (d75 * d75 + d84 * d84) * 0.125f)));
        float c04 = 1.f / (EPS_CURV + sqrtf(fmaf(d04, d04, (d53 * d53 + d62 * d62) * 0.125f)));
        float c08 = 1.f / (EPS_CURV + sqrtf(fmaf(d08, d08, (d71 * d71 + d62 * d62) * 0.125f)));

        float nabla = c02 * fabsf(d02) + c04 * fabsf(d04)
                    + c06 * fabsf(d06) + c08 * fabsf(d08);

        const float* fr0 = &fs[orow * LST];
        const float* fr1 = fr0 + LST;
        const float* fr2 = fr1 + LST;
        float f00 = fr0[cL], f01 = fr0[lx], f02v = fr0[cR];
        float f10 = fr1[cL], f11 = fr1[lx], f12  = fr1[cR];
        float f20 = fr2[cL], f21 = fr2[lx], f22  = fr2[cR];

        float h = 0.f;
        h = fmaf(ul,  w[0], h);  h = fmaf(up,  w[1], h);  h = fmaf(ur,  w[2], h);
        h = fmaf(lf,  w[3], h);  h = fmaf(v,   w[4], h);  h = fmaf(rt,  w[5], h);
        h = fmaf(dl,  w[6], h);  h = fmaf(dn,  w[7], h);  h = fmaf(dr,  w[8], h);
        h = fmaf(f00, w[9], h);  h = fmaf(f01, w[10], h); h = fmaf(f02v, w[11], h);
        h = fmaf(f10, w[12], h); h = fmaf(f11, w[13], h); h = fmaf(f12, w[14], h);
        h = fmaf(f20, w[15], h); h = fmaf(f21, w[16], h); h = fmaf(f22, w[17], h);

        float ham = fmaf(h, scale, shift);
        float t = v + nabla + ham;
        out[(size_t)b * HH * WW + (size_t)gy * WW + gx] = 1.f / (1.f + expf(-t));
    }
}

// ---------------- Host launcher ----------------
extern "C" void kernel_launch(void* const* d_in, const int* in_sizes, int n_in,
                              void* d_out, int out_size, void* d_ws, size_t ws_size,
                              hipStream_t stream) {
    (void)in_sizes; (void)n_in; (void)out_size;
    const float* x     = (const float*)d_in[0];
    const float* f     = (const float*)d_in[1];
    const float* wconv = (const float*)d_in[2];
    const float* gamma = (const float*)d_in[3];
    const float* beta  = (const float*)d_in[4];
    float* out = (float*)d_out;
    float* ws  = (float*)d_ws;

    // Clamp reduction-block count to available workspace (needs 2*nb+2 floats)
    int nb = NB_RED;
    size_t avail = ws_size / sizeof(float);
    while (nb > 1 && ((size_t)2 * nb + 2) > avail) nb >>= 1;

    float* psum = ws;
    float* psqr = ws + nb;
    float* bnp  = ws + 2 * nb;

    hsum_kernel<<<nb, NTHREADS, 0, stream>>>(x, f, wconv, psum, psqr);
    bnstats_kernel<<<1, NTHREADS, 0, stream>>>(psum, psqr, gamma, beta, bnp, nb);

    const int tilesX = (WW + TW - 1) / TW;   // 6
    const int tilesY = (HH + TH - 1) / TH;   // 48
    dim3 grid(tilesX * tilesY, BB);
    fused_kernel<<<grid, NTHREADS, 0, stream>>>(x, f, wconv, bnp, out);
}